// EDM_15040975470940
// MI455X (gfx1250) — compile-verified
//
#include <hip/hip_runtime.h>
#include <hip/hip_bf16.h>

// Euclidean distance matrix on gfx1250 (MI455X).
//   out[b,i,j] = sqrt(max(sq[i] + sq[j] - 2*dot(x[b,i], x[b,j]), 0) + 1e-7)
// Gram matrix computed with V_WMMA_F32_16X16X4_F32 (exact fp32, 16x16 tile per wave32).
// Store-bound: 64 MB output -> ~2.9 us floor @ 23.3 TB/s; x (4 MB) lives in L2.

typedef __attribute__((ext_vector_type(2))) float v2f;
typedef __attribute__((ext_vector_type(8))) float v8f;

#define EDM_B 4
#define EDM_N 2048
#define EDM_D 128
#define EDM_EPS 1e-7f

// ---------------------------------------------------------------------------
// Kernel 1: per-row squared norms. One wave32 per row (B*N = 8192 rows).
// ---------------------------------------------------------------------------
__global__ void edm_sqnorm_kernel(const float* __restrict__ x,
                                  float* __restrict__ sq) {
    const int wave = (blockIdx.x * blockDim.x + threadIdx.x) >> 5;
    const int lane = threadIdx.x & 31;
    if (wave >= EDM_B * EDM_N) return;   // uniform per-wave, grid sized exactly

    const float* row = x + (size_t)wave * EDM_D;
    float s = 0.0f;
#pragma unroll
    for (int k = 0; k < EDM_D; k += 32) {   // 4 coalesced 128B loads per wave
        float v = row[k + lane];
        s = fmaf(v, v, s);
    }
    // wave32 butterfly reduction
#pragma unroll
    for (int off = 16; off > 0; off >>= 1)
        s += __shfl_xor(s, off, 32);
    if (lane == 0) sq[wave] = s;
}

// ---------------------------------------------------------------------------
// Kernel 2: one wave32 per 16x16 output tile.
//   tiles = B * (N/16) * (N/16) = 4 * 128 * 128 = 65536
//   A tile: x[row0..row0+15, :]   (16xD, row-major)
//   B tile: x[col0..col0+15, :]^T (Dx16)  -> same per-lane gather as A
// A layout (f32 16x4): lane<16: VGPR0/1 = K0/K1 ; lane>=16: VGPR0/1 = K2/K3
//   => per lane load float2 at [row0 + (lane&15)]*D + k0 + 2*(lane>>4)
// B layout (f32 4x16, row-striped): mirrored => identical gather with col0.
// C layout: VGPR r -> M = r + 8*(lane>>4), N = lane&15.
// ---------------------------------------------------------------------------
__global__ void edm_tile_kernel(const float* __restrict__ x,
                                const float* __restrict__ sq,
                                float* __restrict__ out) {
    const int lane   = threadIdx.x & 31;
    const int waveId = threadIdx.x >> 5;
    const int tile   = blockIdx.x * (blockDim.x >> 5) + waveId; // 0..65535

    const int b  = tile >> 14;          // / (128*128)
    const int ti = (tile >> 7) & 127;   // row tile
    const int tj = tile & 127;          // col tile
    const int row0 = ti << 4;
    const int col0 = tj << 4;

    const int m    = lane & 15;         // matrix row within tile (A/B), col (C)
    const int koff = (lane >> 4) << 1;  // 0 or 2

    const float* xb = x + (size_t)b * EDM_N * EDM_D;
    const float* ap = xb + (size_t)(row0 + m) * EDM_D + koff;
    const float* bp = xb + (size_t)(col0 + m) * EDM_D + koff;

    v8f acc = {};
#pragma unroll 8
    for (int k0 = 0; k0 < EDM_D; k0 += 4) {
        v2f av = *(const v2f*)(ap + k0);
        v2f bv = *(const v2f*)(bp + k0);
        // D = A(16x4) x B(4x16) + C, exact fp32
        acc = __builtin_amdgcn_wmma_f32_16x16x4_f32(
            /*neg_a=*/false, av, /*neg_b=*/false, bv,
            /*c_mod=*/(short)0, acc, /*reuse_a=*/false, /*reuse_b=*/false);
    }

    // Epilogue: d = sqrt(max(sq_i + sq_j - 2*s, 0) + eps)
    const int   half = lane >> 4;                       // 0 or 1
    const float sqj  = sq[b * EDM_N + col0 + m];        // column norm (N = lane&15)
    const float* sqi = sq + b * EDM_N + row0 + (half << 3);
    float* orow = out + ((size_t)b * EDM_N + row0 + (half << 3)) * EDM_N
                      + col0 + m;
#pragma unroll
    for (int r = 0; r < 8; ++r) {
        float d2 = sqi[r] + sqj - 2.0f * acc[r];
        d2 = fmaxf(d2, 0.0f);
        // written once, never re-read: non-temporal to keep x resident in L2
        __builtin_nontemporal_store(sqrtf(d2 + EDM_EPS), orow + (size_t)r * EDM_N);
    }
}

// ---------------------------------------------------------------------------
extern "C" void kernel_launch(void* const* d_in, const int* in_sizes, int n_in,
                              void* d_out, int out_size, void* d_ws, size_t ws_size,
                              hipStream_t stream) {
    (void)in_sizes; (void)n_in; (void)out_size; (void)ws_size;

    const float* x  = (const float*)d_in[0];
    float* out      = (float*)d_out;
    float* sq       = (float*)d_ws;     // B*N floats = 32 KB scratch

    // Kernel 1: 8192 rows, 1 wave each; 256 thr/block = 8 waves -> 1024 blocks
    edm_sqnorm_kernel<<<1024, 256, 0, stream>>>(x, sq);

    // Kernel 2: 65536 tiles, 1 wave each; 8 waves/block -> 8192 blocks
    edm_tile_kernel<<<8192, 256, 0, stream>>>(x, sq, out);
}